// NumbaHybridBlock_69011534512435
// MI455X (gfx1250) — compile-verified
//
#include <hip/hip_runtime.h>
#include <hip/hip_bf16.h>

typedef __attribute__((ext_vector_type(2))) float v2f;
typedef __attribute__((ext_vector_type(4))) float v4f;
typedef __attribute__((ext_vector_type(8))) float v8f;

#define D_IN  512
#define D_HID 512
#define B_SZ  8
#define T_SZ  2048
#define M_TOT (B_SZ * T_SZ)      // 16384 rows of flattened (B*T, D_IN)

#define KC    64                 // K-chunk staged in LDS
#define LSTR  68                 // padded row stride: 272B = 16B-aligned rows AND
                                 // 4-bank shift/row -> frag reads hit all 64 banks once
#define NROWS (32 + 64 + 64)     // x rows + Wd rows + Wb rows per buffer
#define BUF_FLOATS (NROWS * LSTR)

// ---- async global->LDS copy (CDNA5 ASYNCcnt path), with safe fallback ------
#if defined(__has_builtin)
#if __has_builtin(__builtin_amdgcn_global_load_async_to_lds_b128)
#define HAVE_ASYNC_LDS 1
#endif
#endif

typedef int v4i_gcc __attribute__((vector_size(16)));                 // 4 x i32
typedef __attribute__((address_space(1))) v4i_gcc* glob_v4i_p;        // global
typedef __attribute__((address_space(3))) v4i_gcc* lds_v4i_p;         // LDS

__device__ __forceinline__ void async_copy16(const float* g, float* l) {
#ifdef HAVE_ASYNC_LDS
    __builtin_amdgcn_global_load_async_to_lds_b128(
        (glob_v4i_p)g, (lds_v4i_p)l, /*offset=*/0, /*cpol=*/0);
#else
    *(v4f*)l = *(const v4f*)g;           // synchronous fallback
#endif
}

__device__ __forceinline__ void wait_async_copies() {
#ifdef HAVE_ASYNC_LDS
#if __has_builtin(__builtin_amdgcn_s_wait_asynccnt)
    __builtin_amdgcn_s_wait_asynccnt(0);
#else
    asm volatile("s_wait_asynccnt 0" ::: "memory");
#endif
#endif
}

// ---------------------------------------------------------------------------
// Kernel 1: fused dual-projection GEMM (fp32 WMMA 16x16x4) + CfC gate epilogue.
// Block: 256 threads = 8 waves; tile 32(M) x 64(N); each wave owns a 16x16 tile
// and accumulates BOTH projections (A fragment reused for 2 WMMAs per K-step).
// Double-buffered LDS; chunk k+1 streams in via async-LDS under chunk k WMMAs.
// ---------------------------------------------------------------------------
__global__ __launch_bounds__(256)
void cfc_proj_kernel(const float* __restrict__ x,
                     const float* __restrict__ Wd,
                     const float* __restrict__ bd,
                     const float* __restrict__ Wb,
                     const float* __restrict__ bb,
                     const float* __restrict__ A_log,
                     float* __restrict__ Aout,
                     float* __restrict__ Bout)
{
    extern __shared__ float smem[];      // 2 * BUF_FLOATS

    const int tid  = threadIdx.x;
    const int lane = tid & 31;
    const int wave = tid >> 5;
    const int wm   = wave >> 2;          // 0..1  (M sub-tile)
    const int wn   = wave & 3;           // 0..3  (N sub-tile)
    const int half = lane >> 4;          // 0: K pair {k,k+1}, 1: {k+2,k+3}
    const int l16  = lane & 15;

    const int m0 = blockIdx.x * 32;
    const int n0 = blockIdx.y * 64;

    // per-thread copy assignments (16B each)
    const int xr  = tid >> 4;            // 0..15  (plus +16 per extra pass)
    const int xc4 = (tid & 15) * 4;

    // issue all async copies for one K-chunk into one buffer
    auto issue_chunk = [&](int kc, int bsel) {
        float* bx  = smem + bsel * BUF_FLOATS;
        float* bwd = bx + 32 * LSTR;
        float* bwb = bx + 96 * LSTR;
#pragma unroll
        for (int it = 0; it < 2; ++it) {             // x: 32 rows x 16 float4
            int r = xr + it * 16;
            async_copy16(x + (size_t)(m0 + r) * D_IN + kc + xc4,
                         bx + r * LSTR + xc4);
        }
#pragma unroll
        for (int it = 0; it < 4; ++it) {             // each W: 64 rows x 16 float4
            int r = xr + it * 16;
            async_copy16(Wd + (size_t)(n0 + r) * D_IN + kc + xc4,
                         bwd + r * LSTR + xc4);
            async_copy16(Wb + (size_t)(n0 + r) * D_IN + kc + xc4,
                         bwb + r * LSTR + xc4);
        }
    };

    v8f acc_d = {0.f,0.f,0.f,0.f,0.f,0.f,0.f,0.f};
    v8f acc_b = {0.f,0.f,0.f,0.f,0.f,0.f,0.f,0.f};

    const int NCHUNK = D_IN / KC;        // 8
    issue_chunk(0, 0);
    wait_async_copies();
    __syncthreads();

    for (int c = 0; c < NCHUNK; ++c) {
        const int cur = c & 1;
        if (c + 1 < NCHUNK)
            issue_chunk((c + 1) * KC, (c + 1) & 1);  // overlaps with WMMAs below

        const float* bx  = smem + cur * BUF_FLOATS;
        // fragment bases (ISA 32-bit A 16x4 / B 4x16 layouts):
        // lane&15 = row(A)/col(B); lane>>4 selects K pair.
        const float* ax  = bx + (wm * 16 + l16) * LSTR + half * 2;
        const float* bwd = bx + (32 + wn * 16 + l16) * LSTR + half * 2;
        const float* bwb = bx + (96 + wn * 16 + l16) * LSTR + half * 2;
#pragma unroll
        for (int k = 0; k < KC; k += 4) {
            v2f a   = *(const v2f*)(ax  + k);
            v2f wd2 = *(const v2f*)(bwd + k);
            v2f wb2 = *(const v2f*)(bwb + k);
            acc_d = __builtin_amdgcn_wmma_f32_16x16x4_f32(
                        false, a, false, wd2, (short)0, acc_d, false, false);
            acc_b = __builtin_amdgcn_wmma_f32_16x16x4_f32(
                        false, a, false, wb2, (short)0, acc_b, false, false);
        }
        wait_async_copies();             // my chunk-(c+1) copies landed
        __syncthreads();                 // everyone done reading buf cur & writing next
    }

    // Epilogue. C layout: VGPR i holds rows {i (lanes 0-15), i+8 (lanes 16-31)},
    // col = lane & 15.
    const int col   = n0 + wn * 16 + l16;
    const float bdv = bd[col];
    const float bbv = bb[col];
    const float eA  = __expf(A_log[col]);
#pragma unroll
    for (int i = 0; i < 8; ++i) {
        int row = m0 + wm * 16 + half * 8 + i;
        float dpre  = acc_d[i] + bdv;
        // numerically stable softplus
        float delta = fmaxf(dpre, 0.f) + log1pf(__expf(-fabsf(dpre)));
        float bt    = delta * (acc_b[i] + bbv);
        float at    = __expf(-delta * eA);
        size_t o = (size_t)row * D_HID + col;
        Aout[o] = at;
        Bout[o] = bt;
    }
}

// ---------------------------------------------------------------------------
// Kernel 2: sequential CfC + PLIF scan over T, one thread per (b,d) chain.
// 4096 chains; consecutive lanes = consecutive d => fully coalesced streams.
// 8-step load batching keeps 16 loads in flight ahead of the dependent chain.
// ---------------------------------------------------------------------------
__global__ __launch_bounds__(64)
void cfc_scan_kernel(const float* __restrict__ At,
                     const float* __restrict__ Bt,
                     const float* __restrict__ threshold,
                     const float* __restrict__ log_tau,
                     const float* __restrict__ h0,
                     const float* __restrict__ mem0,
                     float* __restrict__ y_out,
                     float* __restrict__ spk_out,
                     float* __restrict__ m_out)
{
    int gid = blockIdx.x * blockDim.x + threadIdx.x;   // 0..4095
    int b = gid >> 9;          // / D_HID
    int d = gid & (D_HID - 1);

    float tau   = __expf(log_tau[d]);
    tau         = fminf(fmaxf(tau, 0.01f), 1000.f);
    const float decay = __expf(-1.f / tau);
    const float thr   = threshold[d];

    float h = h0[gid];
    float m = mem0[gid];
    const size_t base = (size_t)b * T_SZ * D_HID + d;

    for (int t0 = 0; t0 < T_SZ; t0 += 8) {
        float a[8], bv[8];
#pragma unroll
        for (int j = 0; j < 8; ++j) {
            size_t idx = base + (size_t)(t0 + j) * D_HID;
            a[j]  = At[idx];
            bv[j] = Bt[idx];
        }
#pragma unroll
        for (int j = 0; j < 8; ++j) {
            h = a[j] * h + bv[j];            // CfC linear recurrence
            float y = tanhf(h);              // gfx1250 trans pipe
            m = m * decay + y;               // leaky integrate
            float spk = (m >= thr) ? 1.f : 0.f;
            m -= spk * thr;                  // subtract-on-spike reset
            size_t idx = base + (size_t)(t0 + j) * D_HID;
            y_out[idx]   = y;
            spk_out[idx] = spk;
            m_out[idx]   = m;
        }
    }
}

// ---------------------------------------------------------------------------
extern "C" void kernel_launch(void* const* d_in, const int* in_sizes, int n_in,
                              void* d_out, int out_size, void* d_ws, size_t ws_size,
                              hipStream_t stream)
{
    (void)in_sizes; (void)n_in; (void)out_size; (void)ws_size;

    const float* x     = (const float*)d_in[0];
    const float* Wd    = (const float*)d_in[1];
    const float* bd    = (const float*)d_in[2];
    const float* Wb    = (const float*)d_in[3];
    const float* bb    = (const float*)d_in[4];
    const float* A_log = (const float*)d_in[5];
    const float* thr   = (const float*)d_in[6];
    const float* ltau  = (const float*)d_in[7];
    const float* h0    = (const float*)d_in[8];
    const float* mem0  = (const float*)d_in[9];

    const size_t N = (size_t)M_TOT * D_HID;   // 8,388,608 elements
    float* At = (float*)d_ws;                 // A_t gates (33.5 MB)
    float* Bt = At + N;                       // b_t gates (33.5 MB)

    float* y_out   = (float*)d_out;
    float* spk_out = y_out + N;
    float* m_out   = spk_out + N;

    dim3 grid1(M_TOT / 32, D_HID / 64);       // 512 x 8 blocks
    size_t shmem = 2u * BUF_FLOATS * sizeof(float);   // 87,040 B dynamic LDS
    cfc_proj_kernel<<<grid1, 256, shmem, stream>>>(x, Wd, bd, Wb, bb, A_log, At, Bt);

    cfc_scan_kernel<<<(B_SZ * D_HID) / 64, 64, 0, stream>>>(
        At, Bt, thr, ltau, h0, mem0, y_out, spk_out, m_out);
}